// YOLOLoss_6597069766754
// MI455X (gfx1250) — compile-verified
//
#include <hip/hip_runtime.h>
#include <math.h>
#include <stdint.h>

typedef __attribute__((ext_vector_type(2))) float v2f;
typedef __attribute__((ext_vector_type(8))) float v8f;

#define NB 32   // batch
#define NT 50   // targets per image
#define NC 80   // classes
#define CH 85   // channels per cell

// ANCHORS / 416 (scaled anchors, fractional)
__device__ __constant__ float c_anchW[9] = {
    10.f/416.f, 16.f/416.f, 33.f/416.f, 30.f/416.f, 62.f/416.f,
    59.f/416.f, 116.f/416.f, 156.f/416.f, 373.f/416.f};
__device__ __constant__ float c_anchH[9] = {
    13.f/416.f, 30.f/416.f, 23.f/416.f, 61.f/416.f, 45.f/416.f,
    119.f/416.f, 90.f/416.f, 198.f/416.f, 326.f/416.f};

__device__ __forceinline__ float softplusf(float x) {
    // logaddexp(0, x), numerically stable
    return fmaxf(x, 0.f) + log1pf(expf(-fabsf(x)));
}

// ---------------------------------------------------------------------------
// Wave(32)-sum through the CDNA5 matrix pipe: D = A x ones(4x16) + 0.
// Sum of column 0 of D = sum over ALL A elements = sum over the 32 lanes
// (layout-robust because B is all ones). Column 0 rows 0..7 live in lane 0,
// rows 8..15 in lane 16 (documented 32-bit C/D layout).
// ---------------------------------------------------------------------------
__device__ __forceinline__ float wave_sum_wmma(float x) {
    v2f a; a[0] = x;   a[1] = 0.0f;
    v2f b; b[0] = 1.f; b[1] = 1.f;
    v8f c = {};
    c = __builtin_amdgcn_wmma_f32_16x16x4_f32(
            /*neg_a=*/false, a, /*neg_b=*/false, b,
            /*c_mod=*/(short)0, c, /*reuse_a=*/false, /*reuse_b=*/false);
    float s = c[0]+c[1]+c[2]+c[3]+c[4]+c[5]+c[6]+c[7];
    return __shfl(s, 0, 32) + __shfl(s, 16, 32);
}

// Deterministic block reduction: wmma wave-sum, then thread 0 sums the <=8
// per-wave results in fixed order. Writes exactly one float per block.
__device__ __forceinline__ void block_accum(float v, float* partialOut) {
    __shared__ float sw[8];
    float w = wave_sum_wmma(v);           // all threads execute (EXEC all 1s)
    int lane = threadIdx.x & 31;
    int wid  = threadIdx.x >> 5;
    if (lane == 0) sw[wid] = w;
    __syncthreads();
    if (threadIdx.x == 0) {
        float t = 0.f;
        int nw = (blockDim.x + 31) >> 5;
        for (int k = 0; k < nw; ++k) t += sw[k];
        *partialOut = t;
    }
}

// ---------------------------------------------------------------------------
// Kernel 1: per-target preprocessing (3 scales x 32 batch x 50 targets).
// Float record (8 floats, slots 0..4 used): tx, ty, tw, th, bls
// Dense packed word: gi[7:0] | gj[15:8] | best[17:16] | ign[22:20] | cls[30:24]
// ---------------------------------------------------------------------------
__global__ void target_kernel(const float* __restrict__ targets,
                              float* __restrict__ tgtF,
                              int*   __restrict__ tgtP) {
    int idx = blockIdx.x * blockDim.x + threadIdx.x;
    if (idx >= 3 * NB * NT) return;
    int s = idx / (NB * NT);
    int r = idx % (NB * NT);
    int W = (s == 0) ? 13 : ((s == 1) ? 26 : 52);
    float fW = (float)W;
    const float* t = targets + (size_t)r * 5;
    int   cls = (int)t[0];
    float gx = t[1] * fW, gy = t[2] * fW;
    float gw = t[3] * fW, gh = t[4] * fW;
    int abase = 6 - 3 * s;                 // ANCHORS_MASK[s] = {abase, abase+1, abase+2}
    float area = gw * gh;
    float best_iou = -1.f; int best = 0; int ign = 0;
    for (int a = 0; a < 3; ++a) {
        float aw = c_anchW[abase + a], ah = c_anchH[abase + a];
        float inter = fminf(gw, aw) * fminf(gh, ah);
        float iou = inter / (area + aw * ah - inter + 1e-6f);
        if (iou > best_iou) { best_iou = iou; best = a; }  // argmax, first max wins
        if (iou > 0.5f) ign |= (1 << a);
    }
    int gi = (int)floorf(gx), gj = (int)floorf(gy);
    float* o = tgtF + (size_t)idx * 8;
    o[0] = gx - (float)gi;
    o[1] = gy - (float)gj;
    o[2] = logf(gw / c_anchW[abase + best]);
    o[3] = logf(gh / c_anchH[abase + best]);
    o[4] = 2.0f - t[3] * t[4];
    o[5] = 0.f; o[6] = 0.f; o[7] = 0.f;
    tgtP[idx] = (gi & 0xFF) | ((gj & 0xFF) << 8) | (best << 16) |
                (ign << 20) | ((cls & 0x7F) << 24);
}

// ---------------------------------------------------------------------------
// Kernel 2: one thread per grid cell (b, a, j, i). The whole scale's packed
// target words (NB*NT = 1600 ints, 6.4 KB) are DMA'd into LDS once per block
// via the CDNA5 async-to-LDS path, then the 50-target scan is pure LDS
// broadcast reads. Conf channel is the only global read for unmasked cells.
// ---------------------------------------------------------------------------
__global__ void cell_kernel(const float* __restrict__ pred,
                            const float* __restrict__ tgtF,  // per-scale float records
                            const int*   __restrict__ tgtP,  // per-scale packed words
                            float* __restrict__ partial,     // one per block
                            int W, int H, float balance,
                            float aw0, float aw1, float aw2,
                            float ah0, float ah1, float ah2) {
    __shared__ int sP[NB * NT];   // 6400 B, all batches of this scale

    // --- async stage: global -> LDS, tracked by ASYNCcnt (no VGPR return) ---
    for (int t = threadIdx.x; t < NB * NT; t += blockDim.x) {
        uint32_t ldsAddr = (uint32_t)(uintptr_t)(&sP[t]);  // low 32 bits = LDS offset (ISA 10.2)
        const int* gsrc = tgtP + t;
        asm volatile("global_load_async_to_lds_b32 %0, %1, off"
                     :: "v"(ldsAddr), "v"(gsrc) : "memory");
    }
    asm volatile("s_wait_asynccnt 0" ::: "memory");
    __syncthreads();   // LDS visible to all waves in the block

    int cell = blockIdx.x * blockDim.x + threadIdx.x;
    int total = NB * 3 * H * W;
    float contrib = 0.f;
    if (cell < total) {
        int i = cell % W;
        int j = (cell / W) % H;
        int a = (cell / (W * H)) % 3;
        int b = cell / (W * H * 3);
        const float* p = pred + (size_t)cell * CH;   // layout matches (B,A,H,W,CH)

        float noobj = 1.f; int maskf = 0; int winner = -1;
        unsigned cm0 = 0, cm1 = 0, cm2 = 0;          // class-target bitmask (set semantics)
        const int* tb = &sP[b * NT];
        for (int n = 0; n < NT; ++n) {
            int pk = tb[n];                          // LDS broadcast read
            int gi = pk & 0xFF, gj = (pk >> 8) & 0xFF;
            if (gi == i && gj == j) {
                int best = (pk >> 16) & 0xF;
                int ignb = (pk >> 20) & 0x7;
                if (ignb & (1 << a)) noobj = 0.f;    // noobj .min where iou > thr
                if (best == a) {
                    noobj = 0.f; maskf = 1; winner = n;  // last write wins (scatter .set)
                    int cls = (pk >> 24) & 0x7F;
                    if      (cls < 32) cm0 |= 1u << cls;
                    else if (cls < 64) cm1 |= 1u << (cls - 32);
                    else               cm2 |= 1u << (cls - 64);
                }
            }
        }

        float conf = p[4];
        // mask*bce(conf,1) + noobj*bce(conf,0);  OBJ_RATIO = 5.0
        float closs = (maskf ? (softplusf(conf) - conf) : 0.f) + noobj * softplusf(conf);
        contrib = 5.0f * balance * closs;

        if (maskf) {
            __builtin_prefetch(p, 0, 3);             // global_prefetch_b8: pull full cell vector
            const float* twr = tgtF + (size_t)(b * NT + winner) * 8;
            float tx = twr[0], ty = twr[1], tws = twr[2], ths = twr[3], bls = twr[4];
            float aw = (a == 0) ? aw0 : ((a == 1) ? aw1 : aw2);
            float ah = (a == 0) ? ah0 : ((a == 1) ? ah1 : ah2);
            float dx = 1.f / (1.f + expf(-p[0]));
            float dy = 1.f / (1.f + expf(-p[1]));
            float pw = expf(p[2]) * aw, ph = expf(p[3]) * ah;
            float b1x = dx + (float)i, b1y = dy + (float)j;
            float b2x = tx + (float)i, b2y = ty + (float)j;
            float gw2 = expf(tws) * aw, gh2 = expf(ths) * ah;
            // CIoU
            float b1x1 = b1x - pw * 0.5f, b1y1 = b1y - ph * 0.5f;
            float b1x2 = b1x + pw * 0.5f, b1y2 = b1y + ph * 0.5f;
            float b2x1 = b2x - gw2 * 0.5f, b2y1 = b2y - gh2 * 0.5f;
            float b2x2 = b2x + gw2 * 0.5f, b2y2 = b2y + gh2 * 0.5f;
            float iw = fmaxf(fminf(b1x2, b2x2) - fmaxf(b1x1, b2x1), 0.f);
            float ih = fmaxf(fminf(b1y2, b2y2) - fmaxf(b1y1, b2y1), 0.f);
            float inter = iw * ih;
            float iou = inter / (pw * ph + gw2 * gh2 - inter + 1e-6f);
            float d2 = (b2x - b1x) * (b2x - b1x) + (b2y - b1y) * (b2y - b1y);
            float cw  = fmaxf(b1x2, b2x2) - fminf(b1x1, b2x1);
            float chh = fmaxf(b1y2, b2y2) - fminf(b1y1, b2y1);
            float c2 = cw * cw + chh * chh + 1e-6f;
            float dat = atanf(pw / (ph + 1e-6f)) - atanf(gw2 / (gh2 + 1e-6f));
            float v = 0.405284734569351f * dat * dat;        // 4/pi^2
            float alpha = v / (1.f - iou + v + 1e-6f);
            float ciou = iou - d2 / c2 - alpha * v;
            contrib += 0.05f * (1.f - ciou) * bls;           // BOX_RATIO = 0.05

            // class BCE over 80 channels; CLS_RATIO = 1.0
            float cl = 0.f;
            for (int c = 0; c < NC; ++c) {
                float x = p[5 + c];
                unsigned bit = (c < 32) ? ((cm0 >> c) & 1u)
                             : (c < 64) ? ((cm1 >> (c - 32)) & 1u)
                                        : ((cm2 >> (c - 64)) & 1u);
                cl += softplusf(x) - (bit ? x : 0.f);
            }
            contrib += cl;
        }
    }
    // All threads (including the tail) reach here -> EXEC all 1s at the WMMA.
    block_accum(contrib, &partial[blockIdx.x]);
}

// ---------------------------------------------------------------------------
// Kernel 3: deterministic final reduction, single wave32.
// ---------------------------------------------------------------------------
__global__ void reduce_kernel(const float* __restrict__ partial, int n,
                              float* __restrict__ out) {
    float acc = 0.f;
    for (int k = threadIdx.x; k < n; k += 32) acc += partial[k];
    float tot = wave_sum_wmma(acc);
    if (threadIdx.x == 0) out[0] = tot;
}

extern "C" void kernel_launch(void* const* d_in, const int* in_sizes, int n_in,
                              void* d_out, int out_size, void* d_ws, size_t ws_size,
                              hipStream_t stream) {
    const float* lp = (const float*)d_in[0];
    const float* mp = (const float*)d_in[1];
    const float* sp = (const float*)d_in[2];
    const float* tg = (const float*)d_in[3];
    float* out = (float*)d_out;
    float* ws  = (float*)d_ws;

    float* tgtF    = ws;                               // 3*NB*NT*8 floats
    int*   tgtP    = (int*)(ws + 3 * NB * NT * 8);     // 3*NB*NT ints (dense)
    float* partial = (float*)(tgtP + 3 * NB * NT);     // 1332 block partials

    int ntarget = 3 * NB * NT;                 // 4800
    target_kernel<<<(ntarget + 255) / 256, 256, 0, stream>>>(tg, tgtF, tgtP);

    int blk0 = (NB * 3 * 13 * 13 + 255) / 256; // 64
    int blk1 = (NB * 3 * 26 * 26 + 255) / 256; // 254
    int blk2 = (NB * 3 * 52 * 52 + 255) / 256; // 1014

    // scale 0: anchors 6,7,8  (116,90)(156,198)(373,326)/416, balance 0.4
    cell_kernel<<<blk0, 256, 0, stream>>>(lp, tgtF, tgtP, partial,
        13, 13, 0.4f,
        116.f/416.f, 156.f/416.f, 373.f/416.f,
        90.f/416.f, 198.f/416.f, 326.f/416.f);
    // scale 1: anchors 3,4,5  (30,61)(62,45)(59,119)/416, balance 1.0
    cell_kernel<<<blk1, 256, 0, stream>>>(mp,
        tgtF + (size_t)NB * NT * 8, tgtP + NB * NT, partial + blk0,
        26, 26, 1.0f,
        30.f/416.f, 62.f/416.f, 59.f/416.f,
        61.f/416.f, 45.f/416.f, 119.f/416.f);
    // scale 2: anchors 0,1,2  (10,13)(16,30)(33,23)/416, balance 4.0
    cell_kernel<<<blk2, 256, 0, stream>>>(sp,
        tgtF + (size_t)2 * NB * NT * 8, tgtP + 2 * NB * NT, partial + blk0 + blk1,
        52, 52, 4.0f,
        10.f/416.f, 16.f/416.f, 33.f/416.f,
        13.f/416.f, 30.f/416.f, 23.f/416.f);

    reduce_kernel<<<1, 32, 0, stream>>>(partial, blk0 + blk1 + blk2, out);
}